// AdaptiveWeightedMultiClassCrossEntropyLoss_51934744543274
// MI455X (gfx1250) — compile-verified
//
#include <hip/hip_runtime.h>

typedef float  f4  __attribute__((ext_vector_type(4)));
typedef int    i4  __attribute__((ext_vector_type(4)));
typedef float  v2f __attribute__((ext_vector_type(2)));
typedef float  v8f __attribute__((ext_vector_type(8)));

#define NBINS 8

// ---------------------------------------------------------------- zero ws
__global__ void aw_ce_zero_ws(float* __restrict__ ws) {
    int t = threadIdx.x;
    if (t < 2 * NBINS) ws[t] = 0.0f;
}

// ---------------------------------------------------------------- main reduction
// Streams (B,C=4,S) logits + target + adaptive + mask once (non-temporal b128),
// accumulates class_sum[8] / class_cnt[8] in registers, block-reduces via LDS,
// finishes the 64->16 fold with V_WMMA_F32_16X16X4_F32 (ones-B row-sum),
// then 16 global f32 atomics per block.
__global__ __launch_bounds__(256)
void aw_ce_bins(const float* __restrict__ inp,
                const int*   __restrict__ tgt,
                const int*   __restrict__ atg,
                const float* __restrict__ msk,
                float* __restrict__ ws,
                int num_groups, int sg_shift)
{
    float bsum[NBINS];
    float bcnt[NBINS];
#pragma unroll
    for (int j = 0; j < NBINS; ++j) { bsum[j] = 0.0f; bcnt[j] = 0.0f; }

    const int tid = threadIdx.x;
    const int SG  = 1 << sg_shift;              // groups-of-4 per batch row = S/4

    const f4* __restrict__ inp4 = (const f4*)inp;
    const i4* __restrict__ tgt4 = (const i4*)tgt;
    const i4* __restrict__ atg4 = (const i4*)atg;
    const f4* __restrict__ msk4 = (const f4*)msk;

    const int gstride = gridDim.x * blockDim.x;
    for (int g = blockIdx.x * blockDim.x + tid; g < num_groups; g += gstride) {
        const int b  = g >> sg_shift;
        const int sg = g & (SG - 1);
        const size_t cbase = ((size_t)b * 4) * (size_t)SG + (size_t)sg;

        // 4 channel streams, each contiguous within a batch row -> coalesced b128 NT loads
        f4 x0 = __builtin_nontemporal_load(inp4 + cbase);
        f4 x1 = __builtin_nontemporal_load(inp4 + cbase + (size_t)SG);
        f4 x2 = __builtin_nontemporal_load(inp4 + cbase + 2 * (size_t)SG);
        f4 x3 = __builtin_nontemporal_load(inp4 + cbase + 3 * (size_t)SG);
        i4 t4 = __builtin_nontemporal_load(tgt4 + g);
        i4 a4 = __builtin_nontemporal_load(atg4 + g);
        f4 m4 = __builtin_nontemporal_load(msk4 + g);

#pragma unroll
        for (int i = 0; i < 4; ++i) {
            float va = x0[i], vb = x1[i], vc = x2[i], vd = x3[i];
            float mx  = fmaxf(fmaxf(va, vb), fmaxf(vc, vd));
            float se  = __expf(va - mx) + __expf(vb - mx) +
                        __expf(vc - mx) + __expf(vd - mx);
            float lse = mx + __logf(se);
            int   t   = t4[i];
            float xt  = (t == 0) ? va : (t == 1) ? vb : (t == 2) ? vc : vd;
            float mk  = m4[i];
            float loss  = (lse - xt) * mk;                 // ce * mask
            float valid = (mk > 0.0f) ? 1.0f : 0.0f;
            int   at = a4[i];
            // branchless scatter into 8 private bins: cmp + select + 2 fma per bin
#pragma unroll
            for (int j = 0; j < NBINS; ++j) {
                float f = (at == j) ? 1.0f : 0.0f;
                bsum[j] = fmaf(f, loss,  bsum[j]);
                bcnt[j] = fmaf(f, valid, bcnt[j]);
            }
        }
    }

    // ---- block reduction: 256 x 16 -> 4 x 16 in LDS
    __shared__ float red[256][16];
#pragma unroll
    for (int j = 0; j < NBINS; ++j) {
        red[tid][j]         = bsum[j];
        red[tid][NBINS + j] = bcnt[j];
    }
    __syncthreads();
#pragma unroll
    for (int s = 128; s >= 4; s >>= 1) {
        if (tid < s) {
#pragma unroll
            for (int j = 0; j < 16; ++j) red[tid][j] += red[tid + s][j];
        }
        __syncthreads();
    }

    // ---- final 4x16 -> 16 fold with one f32 WMMA (exact: B is all-ones).
    // A (16x4 f32) layout: lane L holds M=L%16; VGPR0/1 hold K = 2*(L/16)+{0,1}.
    // D (16x16 f32): column N=0 lives in lane 0 (M=0..7) and lane 16 (M=8..15).
    if (tid < 32) {                       // wave 0 only: EXEC all-ones for WMMA
        const int m  = tid & 15;
        const int k0 = (tid >> 4) << 1;   // 0 or 2
        v2f a;  a.x = red[k0][m];  a.y = red[k0 + 1][m];
        v2f bb; bb.x = 1.0f;       bb.y = 1.0f;
        v8f cacc = {0.f, 0.f, 0.f, 0.f, 0.f, 0.f, 0.f, 0.f};
        v8f d = __builtin_amdgcn_wmma_f32_16x16x4_f32(
                    /*neg_a=*/false, a, /*neg_b=*/false, bb,
                    /*c_mod=*/(short)0, cacc, /*reuse_a=*/false, /*reuse_b=*/false);
        if ((tid & 15) == 0) {            // lanes 0 and 16 carry column N=0
            const int base = (tid >> 4) * 8;
#pragma unroll
            for (int r = 0; r < 8; ++r) atomicAdd(&ws[base + r], d[r]);
        }
    }
}

// ---------------------------------------------------------------- finalize (O(8))
__global__ void aw_ce_finalize(const float* __restrict__ ws, float* __restrict__ out,
                               float inv_positions)
{
    if (threadIdx.x != 0 || blockIdx.x != 0) return;
    float s[NBINS], c[NBINS];
    float tot_loss = 0.0f, tot_valid = 0.0f;
    for (int a = 0; a < NBINS; ++a) {
        s[a] = ws[a]; c[a] = ws[NBINS + a];
        tot_loss += s[a]; tot_valid += c[a];
    }
    float fallback = tot_loss * inv_positions;        // mean(loss)
    float cl[NBINS], cc[NBINS];
    float total = 0.0f;
    for (int a = 0; a < NBINS; ++a) {
        bool has = c[a] > 0.0f;
        cl[a] = has ? (s[a] / c[a]) : fallback;
        cc[a] = has ? c[a] : 1.0f;
        total += cl[a] * cc[a];
    }
    float fin = 0.0f;
    for (int a = 0; a < NBINS; ++a) {
        float prop = (total > 0.0f) ? (cl[a] * cc[a] / total) : 0.125f;
        float w = 1.0f + prop;                        // WEIGHT_ALPHA = 1.0
        fin += w * s[a];                              // sum(loss * weights[at])
    }
    out[0] = fin / tot_valid;                         // / sum(mask)  (mask ∈ {0,1})
}

// ---------------------------------------------------------------- launch
extern "C" void kernel_launch(void* const* d_in, const int* in_sizes, int n_in,
                              void* d_out, int out_size, void* d_ws, size_t ws_size,
                              hipStream_t stream)
{
    const float* inp = (const float*)d_in[0];   // (B, 4, S) f32
    const int*   tgt = (const int*)  d_in[1];   // (B, S)
    const int*   atg = (const int*)  d_in[2];   // (B, S)
    const float* msk = (const float*)d_in[3];   // (B, S)
    float* ws  = (float*)d_ws;                  // ws[0..7]=class_sum, ws[8..15]=class_cnt
    float* out = (float*)d_out;

    const int P          = in_sizes[1];         // B*S = 8,388,608
    const int num_groups = P / 4;               // groups of 4 consecutive positions
    const int sg_shift   = 14;                  // log2(S/4), S = 65536 per reference
    (void)n_in; (void)out_size; (void)ws_size;

    aw_ce_zero_ws<<<1, 32, 0, stream>>>(ws);
    aw_ce_bins<<<2048, 256, 0, stream>>>(inp, tgt, atg, msk, ws, num_groups, sg_shift);
    aw_ce_finalize<<<1, 32, 0, stream>>>(ws, out, 1.0f / (float)P);
}